// ModelBLA_36850819400557
// MI455X (gfx1250) — compile-verified
//
#include <hip/hip_runtime.h>
#include <stddef.h>

typedef float v2f __attribute__((ext_vector_type(2)));
typedef float v8f __attribute__((ext_vector_type(8)));

#define NSTEPS 8192
#define NUDIM 4
#define RDIM 128
#define NXDIM 64
#define NYDIM 4
#define CHUNK 64
#define TCH 128 /* NSTEPS / CHUNK */

// Swap lane halves 0-15 <-> 16-31 (wave32): ds_swizzle SWAPX16
// offset[4:0]=and=0x1f, [9:5]=or=0, [14:10]=xor=0x10 -> 0x401f
__device__ __forceinline__ float swapx16(float x) {
  return __int_as_float(__builtin_amdgcn_ds_swizzle(__float_as_int(x), 0x401f));
}

__device__ __forceinline__ v8f wmma4(v2f a, v2f b, v8f c) {
  // D(16x16 f32) = A(16x4 f32) * B(4x16 f32) + C
  return __builtin_amdgcn_wmma_f32_16x16x4_f32(false, a, false, b, (short)0, c,
                                               false, false);
}

// nd[m] += A(64x64, frags Af[m][kk]) * X(64x16 in B-operand frags xb0/xb1)
__device__ __forceinline__ void mm_accum(const v2f Af[4][16],
                                         const float xb0[16],
                                         const float xb1[16], v8f nd[4]) {
#pragma unroll
  for (int m = 0; m < 4; ++m) {
    v8f acc = nd[m];
#pragma unroll
    for (int kk = 0; kk < 16; ++kk) {
      v2f b;
      b.x = xb0[kk];
      b.y = xb1[kk];
      acc = wmma4(Af[m][kk], b, acc);
    }
    nd[m] = acc;
  }
}

// Convert WMMA C/D-layout state (nd[4] covering rows 16m+j / 16m+8+j) into
// B-operand layout (xb0[k]: rows 4k | 4k+2 in lane halves; xb1[k]: 4k+1|4k+3)
__device__ __forceinline__ void d_to_b(const v8f nd[4], float xb0[16],
                                       float xb1[16], bool lo) {
#pragma unroll
  for (int m = 0; m < 4; ++m) {
    float s0 = swapx16(nd[m][0]);
    float s1 = swapx16(nd[m][1]);
    float s2 = swapx16(nd[m][2]);
    float s3 = swapx16(nd[m][3]);
    float s4 = swapx16(nd[m][4]);
    float s5 = swapx16(nd[m][5]);
    float s6 = swapx16(nd[m][6]);
    float s7 = swapx16(nd[m][7]);
    xb0[4 * m + 0] = lo ? nd[m][0] : s2;  // rows 0|2   (+16m)
    xb1[4 * m + 0] = lo ? nd[m][1] : s3;  // rows 1|3
    xb0[4 * m + 1] = lo ? nd[m][4] : s6;  // rows 4|6
    xb1[4 * m + 1] = lo ? nd[m][5] : s7;  // rows 5|7
    xb0[4 * m + 2] = lo ? s0 : nd[m][2];  // rows 8|10
    xb1[4 * m + 2] = lo ? s1 : nd[m][3];  // rows 9|11
    xb0[4 * m + 3] = lo ? s4 : nd[m][6];  // rows 12|14
    xb1[4 * m + 3] = lo ? s5 : nd[m][7];  // rows 13|15
  }
}

// Load a 64x64 row-major matrix into A-operand fragments.
__device__ __forceinline__ void load_a64(const float* __restrict__ M,
                                         v2f Af[4][16], int r, int off) {
#pragma unroll
  for (int m = 0; m < 4; ++m)
#pragma unroll
    for (int kk = 0; kk < 16; ++kk) {
      const float* p = M + (16 * m + r) * 64 + 4 * kk + off;
      v2f a;
      a.x = p[0];
      a.y = p[1];
      Af[m][kk] = a;
    }
}

// Load B_u (64x4) into 4 A-operand fragments (one per M tile).
__device__ __forceinline__ void load_bu(const float* __restrict__ B,
                                        v2f Bu[4], int r, int off) {
#pragma unroll
  for (int m = 0; m < 4; ++m) {
    const float* p = B + (16 * m + r) * 4 + off;
    v2f a;
    a.x = p[0];
    a.y = p[1];
    Bu[m] = a;
  }
}

// ---------------- Phase 0: tiny matrix squaring (A^2 ... A^64) -------------
__global__ __launch_bounds__(256) void matsq_kernel(
    const float* __restrict__ in, float* __restrict__ out) {
  int idx = blockIdx.x * blockDim.x + threadIdx.x;  // 0..4095
  int row = idx >> 6, c = idx & 63;
  float s = 0.f;
  for (int k = 0; k < 64; ++k) s = fmaf(in[row * 64 + k], in[k * 64 + c], s);
  out[idx] = s;
}

// ---------------- Phase 1: per-chunk partial (zero-start) ------------------
__global__ __launch_bounds__(256) void phase1_partial(
    const float* __restrict__ u, const float* __restrict__ A,
    const float* __restrict__ B_u, float* __restrict__ P) {
  const int lane = threadIdx.x & 31;
  const int wave = threadIdx.x >> 5;  // column slice 0..7
  const int t = blockIdx.x;           // chunk
  const bool lo = lane < 16;
  const int r = lane & 15;
  const int off = lo ? 0 : 2;
  const int col = wave * 16 + r;
  const int r0 = lo ? 0 : 2, r1 = lo ? 1 : 3;

  v2f Af[4][16];
  load_a64(A, Af, r, off);
  v2f Bu[4];
  load_bu(B_u, Bu, r, off);

  float xb0[16], xb1[16];
#pragma unroll
  for (int kk = 0; kk < 16; ++kk) {
    xb0[kk] = 0.f;
    xb1[kk] = 0.f;
  }

  const float* ub = u + (size_t)t * CHUNK * NUDIM * RDIM;
  for (int j = 0; j < CHUNK; ++j) {
    const float* up = ub + j * NUDIM * RDIM;
    v2f uo;
    uo.x = up[r0 * RDIM + col];
    uo.y = up[r1 * RDIM + col];
    v8f nd[4];
    v8f z = {0.f, 0.f, 0.f, 0.f, 0.f, 0.f, 0.f, 0.f};
#pragma unroll
    for (int m = 0; m < 4; ++m) nd[m] = wmma4(Bu[m], uo, z);
    mm_accum(Af, xb0, xb1, nd);
    d_to_b(nd, xb0, xb1, lo);
  }

  float* Pt = P + (size_t)t * NXDIM * RDIM;
#pragma unroll
  for (int kk = 0; kk < 16; ++kk) {
    int ra = 4 * kk + off;
    Pt[ra * RDIM + col] = xb0[kk];
    Pt[(ra + 1) * RDIM + col] = xb1[kk];
  }
}

// ---------------- Phase 2: scan over chunks  S_{t+1} = A^L S_t + P_t -------
__global__ __launch_bounds__(256) void phase2_scan(
    const float* __restrict__ AL, const float* __restrict__ P,
    float* __restrict__ S) {
  const int lane = threadIdx.x & 31;
  const int wave = threadIdx.x >> 5;
  const bool lo = lane < 16;
  const int r = lane & 15;
  const int off = lo ? 0 : 2;
  const int col = wave * 16 + r;
  const int rhalf = lo ? 0 : 8;

  v2f Af[4][16];
  load_a64(AL, Af, r, off);

  float xb0[16], xb1[16];
#pragma unroll
  for (int kk = 0; kk < 16; ++kk) {
    xb0[kk] = 0.f;
    xb1[kk] = 0.f;
  }

  for (int t = 0; t < TCH; ++t) {
    // store S_t (chunk-start state)
    float* St = S + (size_t)t * NXDIM * RDIM;
#pragma unroll
    for (int kk = 0; kk < 16; ++kk) {
      int ra = 4 * kk + off;
      St[ra * RDIM + col] = xb0[kk];
      St[(ra + 1) * RDIM + col] = xb1[kk];
    }
    // acc := P_t  (loaded directly in C/D layout)
    const float* Pt = P + (size_t)t * NXDIM * RDIM;
    v8f nd[4];
#pragma unroll
    for (int m = 0; m < 4; ++m)
#pragma unroll
      for (int j = 0; j < 8; ++j)
        nd[m][j] = Pt[(16 * m + rhalf + j) * RDIM + col];
    // acc += A^L * S_t ; next state
    mm_accum(Af, xb0, xb1, nd);
    d_to_b(nd, xb0, xb1, lo);
  }
}

// ---------------- Phase 3: replay chunks, emit X_k and Y_k -----------------
__global__ __launch_bounds__(256) void phase3_emit(
    const float* __restrict__ u, const float* __restrict__ A,
    const float* __restrict__ B_u, const float* __restrict__ C_y,
    const float* __restrict__ D_yu, const float* __restrict__ S,
    float* __restrict__ Yout, float* __restrict__ Xout) {
  const int lane = threadIdx.x & 31;
  const int wave = threadIdx.x >> 5;
  const int t = blockIdx.x;
  const bool lo = lane < 16;
  const int r = lane & 15;
  const int off = lo ? 0 : 2;
  const int col = wave * 16 + r;
  const int r0 = lo ? 0 : 2, r1 = lo ? 1 : 3;

  v2f Af[4][16];
  load_a64(A, Af, r, off);
  v2f Bu[4];
  load_bu(B_u, Bu, r, off);

  // C_y (4x64) and D_yu (4x4) zero-padded to 16 rows, in A-operand fragments
  v2f Cy[16];
#pragma unroll
  for (int kk = 0; kk < 16; ++kk) {
    v2f c = {0.f, 0.f};
    if (r < NYDIM) {
      const float* p = C_y + r * 64 + 4 * kk + off;
      c.x = p[0];
      c.y = p[1];
    }
    Cy[kk] = c;
  }
  v2f Dy = {0.f, 0.f};
  if (r < NYDIM) {
    const float* p = D_yu + r * 4 + off;
    Dy.x = p[0];
    Dy.y = p[1];
  }

  // load chunk-start state S_t into B-operand layout
  const float* St = S + (size_t)t * NXDIM * RDIM;
  float xb0[16], xb1[16];
#pragma unroll
  for (int kk = 0; kk < 16; ++kk) {
    int ra = 4 * kk + off;
    xb0[kk] = St[ra * RDIM + col];
    xb1[kk] = St[(ra + 1) * RDIM + col];
  }

  const float* ub = u + (size_t)t * CHUNK * NUDIM * RDIM;
  for (int j = 0; j < CHUNK; ++j) {
    const size_t n = (size_t)t * CHUNK + j;
    // ---- record X_n (pre-update state) ----
    float* Xn = Xout + n * NXDIM * RDIM;
#pragma unroll
    for (int kk = 0; kk < 16; ++kk) {
      int ra = 4 * kk + off;
      Xn[ra * RDIM + col] = xb0[kk];
      Xn[(ra + 1) * RDIM + col] = xb1[kk];
    }
    // ---- load U_n as B operand ----
    const float* up = ub + j * NUDIM * RDIM;
    v2f uo;
    uo.x = up[r0 * RDIM + col];
    uo.y = up[r1 * RDIM + col];
    // ---- Y_n = Cy*X_n + Dy*U_n (rows 0..3 of a padded 16x16 WMMA) ----
    v8f z = {0.f, 0.f, 0.f, 0.f, 0.f, 0.f, 0.f, 0.f};
    v8f yacc = wmma4(Dy, uo, z);
#pragma unroll
    for (int kk = 0; kk < 16; ++kk) {
      v2f b;
      b.x = xb0[kk];
      b.y = xb1[kk];
      yacc = wmma4(Cy[kk], b, yacc);
    }
    if (lo) {
      float* Yn = Yout + n * NYDIM * RDIM;
      Yn[0 * RDIM + col] = yacc[0];
      Yn[1 * RDIM + col] = yacc[1];
      Yn[2 * RDIM + col] = yacc[2];
      Yn[3 * RDIM + col] = yacc[3];
    }
    // ---- X_{n+1} = A*X_n + Bu*U_n ----
    v8f nd[4];
#pragma unroll
    for (int m = 0; m < 4; ++m) nd[m] = wmma4(Bu[m], uo, z);
    mm_accum(Af, xb0, xb1, nd);
    d_to_b(nd, xb0, xb1, lo);
  }
}

extern "C" void kernel_launch(void* const* d_in, const int* in_sizes, int n_in,
                              void* d_out, int out_size, void* d_ws,
                              size_t ws_size, hipStream_t stream) {
  (void)in_sizes;
  (void)n_in;
  (void)out_size;
  (void)ws_size;
  const float* u = (const float*)d_in[0];
  const float* A = (const float*)d_in[1];
  const float* B_u = (const float*)d_in[2];
  const float* C_y = (const float*)d_in[3];
  const float* D_yu = (const float*)d_in[4];

  float* Y = (float*)d_out;                                    // 8192*4*128
  float* X = (float*)d_out + (size_t)NSTEPS * NYDIM * RDIM;    // 8192*64*128

  float* ws = (float*)d_ws;
  float* P = ws;                       // TCH*64*128 = 1,048,576 floats
  float* S = ws + (size_t)1048576;     // 1,048,576 floats
  float* W0 = ws + (size_t)2097152;    // 4096 floats
  float* W1 = W0 + 4096;               // 4096 floats

  // A^64 via repeated squaring: A->W0(A^2)->W1(A^4)->W0(A^8)->W1(A^16)
  // ->W0(A^32)->W1(A^64)
  matsq_kernel<<<16, 256, 0, stream>>>(A, W0);
  matsq_kernel<<<16, 256, 0, stream>>>(W0, W1);
  matsq_kernel<<<16, 256, 0, stream>>>(W1, W0);
  matsq_kernel<<<16, 256, 0, stream>>>(W0, W1);
  matsq_kernel<<<16, 256, 0, stream>>>(W1, W0);
  matsq_kernel<<<16, 256, 0, stream>>>(W0, W1);

  phase1_partial<<<TCH, 256, 0, stream>>>(u, A, B_u, P);
  phase2_scan<<<1, 256, 0, stream>>>(W1, P, S);
  phase3_emit<<<TCH, 256, 0, stream>>>(u, A, B_u, C_y, D_yu, S, Y, X);
}